// LAMP_2731599201055
// MI455X (gfx1250) — compile-verified
//
#include <hip/hip_runtime.h>
#include <math.h>
#include <stdint.h>

// MI455X / gfx1250: wave32, WMMA f32 16x16x4, async global->LDS staging.
typedef __attribute__((ext_vector_type(2))) float v2f;
typedef __attribute__((ext_vector_type(8))) float v8f;

#define B_BATCH 8192
#define NFREQ   48
#define NTIME   512
#define NLAYER  8
#define KC      64                 // K-chunk staged per async round
#define NCHUNK  (NTIME / KC)       // 8

__device__ __forceinline__ v8f wmma_f32(v2f a, v2f b, v8f c) {
  // D = A(16x4) x B(4x16) + C(16x16), fp32, wave32
  return __builtin_amdgcn_wmma_f32_16x16x4_f32(false, a, false, b, (short)0, c,
                                               false, false);
}

__device__ __forceinline__ float block_reduce_256(float v, float* red) {
  const int t = threadIdx.x;
  red[t] = v;
  __syncthreads();
#pragma unroll
  for (int s = 128; s > 0; s >>= 1) {
    if (t < s) red[t] += red[t + s];
    __syncthreads();
  }
  float r = red[0];
  __syncthreads();
  return r;
}

// Async-stage one K-chunk (KC x 48 floats) of W_re and W_im into LDS.
// 256 threads x 3 iters x 16B = 12288B per matrix. ASYNCcnt += 6 per wave.
// Flat-aperture rule: low 32 bits of a generic __shared__ pointer are the
// LDS byte address, which is what VDST of global_load_async_to_lds expects.
__device__ __forceinline__ void stage_w_chunk(const float* __restrict__ Wre,
                                              const float* __restrict__ Wim,
                                              float* sbuf, int c) {
  const int t = threadIdx.x;
  const float* gre = Wre + c * (KC * NFREQ);
  const float* gim = Wim + c * (KC * NFREQ);
  float* sre = sbuf;
  float* sim = sbuf + KC * NFREQ;
#pragma unroll
  for (int i = 0; i < 3; ++i) {
    const int off = t * 4 + i * 1024;  // in floats; 4 floats = 16 bytes
    const uint32_t lre = (uint32_t)(uintptr_t)(sre + off);
    const uint32_t lim = (uint32_t)(uintptr_t)(sim + off);
    asm volatile("global_load_async_to_lds_b128 %0, %1, off"
                 :: "v"(lre), "v"(gre + off)
                 : "memory");
    asm volatile("global_load_async_to_lds_b128 %0, %1, off"
                 :: "v"(lim), "v"(gim + off)
                 : "memory");
  }
}

// ---------------------------------------------------------------------------
// Split u(B,48,2) into z_re/z_im and emit per-block partial sums of |z|^2.
// grid = B*48/256 = 1536 blocks, 256 threads.
__global__ void k_init(const float* __restrict__ u, float* __restrict__ zre,
                       float* __restrict__ zim, float* __restrict__ psig) {
  __shared__ float red[256];
  const int i = blockIdx.x * 256 + threadIdx.x;  // i < B*48
  const float re = u[2 * i + 0];
  const float im = u[2 * i + 1];
  zre[i] = re;
  zim[i] = im;
  const float s = block_reduce_256(re * re + im * im, red);
  if (threadIdx.x == 0) psig[blockIdx.x] = s;
}

// sigma = sqrt(sum partials) / sqrt(48)  -> scal[0].  1 block x 256.
__global__ void k_reduce_sigma(const float* __restrict__ p, int n,
                               float* __restrict__ scal) {
  __shared__ float red[256];
  float s = 0.f;
  for (int i = threadIdx.x; i < n; i += 256) s += p[i];
  s = block_reduce_256(s, red);
  if (threadIdx.x == 0) scal[0] = sqrtf(s) / sqrtf(48.0f);
}

// b_re = sum(pbre)/48 -> scal[1]; b_im -> scal[2].  1 block x 256.
__global__ void k_reduce_b(const float* __restrict__ pr,
                           const float* __restrict__ pi, int n,
                           float* __restrict__ scal) {
  __shared__ float red[256];
  float sr = 0.f, si = 0.f;
  for (int i = threadIdx.x; i < n; i += 256) {
    sr += pr[i];
    si += pi[i];
  }
  sr = block_reduce_256(sr, red);
  si = block_reduce_256(si, red);
  if (threadIdx.x == 0) {
    scal[1] = sr / 48.0f;
    scal[2] = si / 48.0f;
  }
}

// ---------------------------------------------------------------------------
// Forward: Z = z @ W^T (complex), R = H + Z, soft-threshold -> Hn (stored
// into H in place), per-block partial sums of d.  W = A_re/A_im (512x48).
// grid = B/16 = 512 blocks x 256 threads (8 waves).  Wave w covers N-tiles
// {w, w+8, w+16, w+24}; block covers one 16-row M-tile.  B-fragments here are
// contiguous b64 loads with no intra-block redundancy -> direct global.
__global__ void k_forward(const float* __restrict__ zre,
                          const float* __restrict__ zim,
                          const float* __restrict__ Wre,
                          const float* __restrict__ Wim,
                          const float* __restrict__ theta_l,
                          float* __restrict__ Hre, float* __restrict__ Him,
                          const float* __restrict__ scal,
                          float* __restrict__ pbre, float* __restrict__ pbim,
                          int first) {
  __shared__ float red[256];
  const int lane = threadIdx.x & 31;
  const int wave = threadIdx.x >> 5;
  const int m0 = blockIdx.x * 16;

  const float sigma = scal[0];
  const float th0 = theta_l[0];
  const float th1 = theta_l[1];
  const float lam = th0 * sigma;

  // A fragments: z tile 16x48 -> 12 k-steps of 16x4 (re and im).
  // ISA layout: lane m = lane&15; VGPR0 holds K = base, VGPR1 K = base+1,
  // base = (lane<16) ? 0 : 2.
  v2f are[12], aim[12];
  {
    const int m = m0 + (lane & 15);
    const int kbase = (lane & 16) ? 2 : 0;
    const float* pr = zre + m * NFREQ + kbase;
    const float* pi = zim + m * NFREQ + kbase;
#pragma unroll
    for (int ks = 0; ks < 12; ++ks) {
      are[ks].x = pr[ks * 4 + 0];
      are[ks].y = pr[ks * 4 + 1];
      aim[ks].x = pi[ks * 4 + 0];
      aim[ks].y = pi[ks * 4 + 1];
    }
  }

  float dsum_re = 0.f, dsum_im = 0.f;

  for (int t = 0; t < 4; ++t) {
    const int n0 = (wave + t * 8) * 16;
    v8f accre = {};
    v8f accim = {};
    const int n = n0 + (lane & 15);
    const int kbase = (lane & 16) ? 2 : 0;
#pragma unroll
    for (int ks = 0; ks < 12; ++ks) {
      const int k = ks * 4 + kbase;
      // B operand is W^T, i.e. B[k][n] = W[n][k]  (W row-major 512x48)
      v2f bre, bim, nbim;
      bre.x = Wre[n * NFREQ + k];
      bre.y = Wre[n * NFREQ + k + 1];
      bim.x = Wim[n * NFREQ + k];
      bim.y = Wim[n * NFREQ + k + 1];
      nbim.x = -bim.x;
      nbim.y = -bim.y;
      accre = wmma_f32(are[ks], bre, accre);   // + z_re * Wre^T
      accre = wmma_f32(aim[ks], nbim, accre);  // - z_im * Wim^T
      accim = wmma_f32(are[ks], bim, accim);   // + z_re * Wim^T
      accim = wmma_f32(aim[ks], bre, accim);   // + z_im * Wre^T
    }
    // Epilogue: C/D layout -> row = m0 + i + (lane>=16 ? 8 : 0), col = n0+lane%16
#pragma unroll
    for (int i = 0; i < 8; ++i) {
      const int row = m0 + i + ((lane & 16) ? 8 : 0);
      const int col = n0 + (lane & 15);
      const int idx = row * NTIME + col;
      const float Rre = accre[i] + (first ? 0.f : Hre[idx]);
      const float Rim = accim[i] + (first ? 0.f : Him[idx]);
      const float mr = fabsf(Rre);
      const float mi = fabsf(Rim);
      const float shr = fmaxf(mr - lam, 0.f);
      const float shi = fmaxf(mi - lam, 0.f);
      const float sgr = (Rre > 0.f) ? 1.f : ((Rre < 0.f) ? -1.f : 0.f);
      const float sgi = (Rim > 0.f) ? 1.f : ((Rim < 0.f) ? -1.f : 0.f);
      Hre[idx] = th1 * sgr * shr;
      Him[idx] = th1 * sgi * shi;
      dsum_re += (mr > lam) ? th1 : 0.f;
      dsum_im += (mi > lam) ? th1 : 0.f;
    }
  }

  const float br = block_reduce_256(dsum_re, red);
  const float bi = block_reduce_256(dsum_im, red);
  if (threadIdx.x == 0) {
    pbre[blockIdx.x] = br;
    pbim[blockIdx.x] = bi;
  }
}

// ---------------------------------------------------------------------------
// Backward: h = Hn @ W (complex, K=512), z' = u - h + b*z (in place), plus
// per-block partial sums of |z'|^2 for the next layer's sigma.
// grid = B/(16*8) = 64 blocks x 256 threads; wave -> one 16-row M-tile,
// covering all 3 N-tiles (48 cols).  W is staged into LDS in double-buffered
// K-chunks via async global->LDS (all 8 waves share the same B-fragments).
__global__ void k_backward(const float* __restrict__ Hre,
                           const float* __restrict__ Him,
                           const float* __restrict__ Wre,
                           const float* __restrict__ Wim,
                           const float* __restrict__ u,
                           float* __restrict__ zre, float* __restrict__ zim,
                           const float* __restrict__ scal,
                           float* __restrict__ psig) {
  __shared__ float red[256];
  __shared__ __align__(16) float sW[2][2][KC * NFREQ];  // [buf][re/im][k*48+n]
  const int lane = threadIdx.x & 31;
  const int wave = threadIdx.x >> 5;
  const int m0 = (blockIdx.x * 8 + wave) * 16;

  const float b_re = scal[1];
  const float b_im = scal[2];

  stage_w_chunk(Wre, Wim, &sW[0][0][0], 0);
  stage_w_chunk(Wre, Wim, &sW[1][0][0], 1);

  v8f ar[3], ai[3];
  {
    v8f zv = {};
#pragma unroll
    for (int nt = 0; nt < 3; ++nt) {
      ar[nt] = zv;
      ai[nt] = zv;
    }
  }

  const int mrow = m0 + (lane & 15);
  const int kbase = (lane & 16) ? 2 : 0;

  for (int c = 0; c < NCHUNK; ++c) {
    if (c + 1 < NCHUNK) {
      asm volatile("s_wait_asynccnt 6" ::: "memory");  // chunk c landed
    } else {
      asm volatile("s_wait_asynccnt 0" ::: "memory");
    }
    __syncthreads();
    const float* sre = &sW[c & 1][0][0];
    const float* sim = &sW[c & 1][1][0];
    for (int ks = 0; ks < KC / 4; ++ks) {
      const int kl = ks * 4 + kbase;
      const int kg = c * KC + kl;
      v2f a_re, a_im;
      a_re.x = Hre[mrow * NTIME + kg];
      a_re.y = Hre[mrow * NTIME + kg + 1];
      a_im.x = Him[mrow * NTIME + kg];
      a_im.y = Him[mrow * NTIME + kg + 1];
#pragma unroll
      for (int nt = 0; nt < 3; ++nt) {
        const int n = nt * 16 + (lane & 15);
        v2f bre, bim, nbim;
        bre.x = sre[kl * NFREQ + n];
        bre.y = sre[(kl + 1) * NFREQ + n];
        bim.x = sim[kl * NFREQ + n];
        bim.y = sim[(kl + 1) * NFREQ + n];
        nbim.x = -bim.x;
        nbim.y = -bim.y;
        ar[nt] = wmma_f32(a_re, bre, ar[nt]);
        ar[nt] = wmma_f32(a_im, nbim, ar[nt]);
        ai[nt] = wmma_f32(a_re, bim, ai[nt]);
        ai[nt] = wmma_f32(a_im, bre, ai[nt]);
      }
    }
    __syncthreads();
    if (c + 2 < NCHUNK) stage_w_chunk(Wre, Wim, &sW[c & 1][0][0], c + 2);
  }

  float ssum = 0.f;
#pragma unroll
  for (int nt = 0; nt < 3; ++nt) {
#pragma unroll
    for (int i = 0; i < 8; ++i) {
      const int row = m0 + i + ((lane & 16) ? 8 : 0);
      const int col = nt * 16 + (lane & 15);
      const int idx = row * NFREQ + col;
      const float zr = zre[idx];
      const float zi2 = zim[idx];
      const float ur = u[2 * idx + 0];
      const float ui = u[2 * idx + 1];
      const float znr = ur - ar[nt][i] + b_re * zr;
      const float zni = ui - ai[nt][i] + b_im * zi2;
      zre[idx] = znr;
      zim[idx] = zni;
      ssum += znr * znr + zni * zni;
    }
  }
  ssum = block_reduce_256(ssum, red);
  if (threadIdx.x == 0) psig[blockIdx.x] = ssum;
}

// ---------------------------------------------------------------------------
// Final: h = H @ DFT (complex), write interleaved (B,48,2) output.
// Same decomposition + async W staging as k_backward.
__global__ void k_final(const float* __restrict__ Hre,
                        const float* __restrict__ Him,
                        const float* __restrict__ Dre,
                        const float* __restrict__ Dim,
                        float* __restrict__ out) {
  __shared__ __align__(16) float sW[2][2][KC * NFREQ];
  const int lane = threadIdx.x & 31;
  const int wave = threadIdx.x >> 5;
  const int m0 = (blockIdx.x * 8 + wave) * 16;

  stage_w_chunk(Dre, Dim, &sW[0][0][0], 0);
  stage_w_chunk(Dre, Dim, &sW[1][0][0], 1);

  v8f ar[3], ai[3];
  {
    v8f zv = {};
#pragma unroll
    for (int nt = 0; nt < 3; ++nt) {
      ar[nt] = zv;
      ai[nt] = zv;
    }
  }

  const int mrow = m0 + (lane & 15);
  const int kbase = (lane & 16) ? 2 : 0;

  for (int c = 0; c < NCHUNK; ++c) {
    if (c + 1 < NCHUNK) {
      asm volatile("s_wait_asynccnt 6" ::: "memory");
    } else {
      asm volatile("s_wait_asynccnt 0" ::: "memory");
    }
    __syncthreads();
    const float* sre = &sW[c & 1][0][0];
    const float* sim = &sW[c & 1][1][0];
    for (int ks = 0; ks < KC / 4; ++ks) {
      const int kl = ks * 4 + kbase;
      const int kg = c * KC + kl;
      v2f a_re, a_im;
      a_re.x = Hre[mrow * NTIME + kg];
      a_re.y = Hre[mrow * NTIME + kg + 1];
      a_im.x = Him[mrow * NTIME + kg];
      a_im.y = Him[mrow * NTIME + kg + 1];
#pragma unroll
      for (int nt = 0; nt < 3; ++nt) {
        const int n = nt * 16 + (lane & 15);
        v2f bre, bim, nbim;
        bre.x = sre[kl * NFREQ + n];
        bre.y = sre[(kl + 1) * NFREQ + n];
        bim.x = sim[kl * NFREQ + n];
        bim.y = sim[(kl + 1) * NFREQ + n];
        nbim.x = -bim.x;
        nbim.y = -bim.y;
        ar[nt] = wmma_f32(a_re, bre, ar[nt]);   // + H_re*D_re
        ar[nt] = wmma_f32(a_im, nbim, ar[nt]);  // - H_im*D_im
        ai[nt] = wmma_f32(a_re, bim, ai[nt]);   // + H_re*D_im
        ai[nt] = wmma_f32(a_im, bre, ai[nt]);   // + H_im*D_re
      }
    }
    __syncthreads();
    if (c + 2 < NCHUNK) stage_w_chunk(Dre, Dim, &sW[c & 1][0][0], c + 2);
  }

#pragma unroll
  for (int nt = 0; nt < 3; ++nt) {
#pragma unroll
    for (int i = 0; i < 8; ++i) {
      const int row = m0 + i + ((lane & 16) ? 8 : 0);
      const int col = nt * 16 + (lane & 15);
      const int idx = row * NFREQ + col;
      out[2 * idx + 0] = ar[nt][i];
      out[2 * idx + 1] = ai[nt][i];
    }
  }
}

// ---------------------------------------------------------------------------
extern "C" void kernel_launch(void* const* d_in, const int* in_sizes, int n_in,
                              void* d_out, int out_size, void* d_ws,
                              size_t ws_size, hipStream_t stream) {
  (void)in_sizes;
  (void)n_in;
  (void)out_size;
  (void)ws_size;  // needs ~37 MB; harness scratch assumed sufficient

  const float* u     = (const float*)d_in[0];  // (8192,48,2)
  const float* A_re  = (const float*)d_in[1];  // (8,512,48)
  const float* A_im  = (const float*)d_in[2];  // (8,512,48)
  const float* theta = (const float*)d_in[3];  // (8,3)
  const float* D_re  = (const float*)d_in[4];  // (512,48)
  const float* D_im  = (const float*)d_in[5];  // (512,48)
  float* out = (float*)d_out;                  // (8192,48,2)

  const int BK = B_BATCH * NFREQ;   // 393216
  const int BT = B_BATCH * NTIME;   // 4194304
  float* w    = (float*)d_ws;
  float* zre  = w;
  float* zim  = zre + BK;
  float* Hre  = zim + BK;
  float* Him  = Hre + BT;
  float* psig = Him + BT;           // 1536 partials (max)
  float* pbre = psig + 1536;        // 512 partials
  float* pbim = pbre + 512;         // 512 partials
  float* scal = pbim + 512;         // [sigma, b_re, b_im]

  k_init<<<BK / 256, 256, 0, stream>>>(u, zre, zim, psig);
  k_reduce_sigma<<<1, 256, 0, stream>>>(psig, BK / 256, scal);

  for (int l = 0; l < NLAYER; ++l) {
    const float* Wre = A_re + l * NTIME * NFREQ;
    const float* Wim = A_im + l * NTIME * NFREQ;
    k_forward<<<B_BATCH / 16, 256, 0, stream>>>(zre, zim, Wre, Wim,
                                                theta + l * 3, Hre, Him, scal,
                                                pbre, pbim, l == 0 ? 1 : 0);
    k_reduce_b<<<1, 256, 0, stream>>>(pbre, pbim, B_BATCH / 16, scal);
    k_backward<<<B_BATCH / 128, 256, 0, stream>>>(Hre, Him, Wre, Wim, u, zre,
                                                  zim, scal, psig);
    if (l < NLAYER - 1)
      k_reduce_sigma<<<1, 256, 0, stream>>>(psig, B_BATCH / 128, scal);
  }

  k_final<<<B_BATCH / 128, 256, 0, stream>>>(Hre, Him, D_re, D_im, out);
}